// GNN_81647328297535
// MI455X (gfx1250) — compile-verified
//
#include <hip/hip_runtime.h>
#include <hip/hip_bf16.h>

#define BATCH 2
#define NN 1024
#define HD 8
#define NNODES (BATCH * NN)

typedef __attribute__((ext_vector_type(2))) float v2f;
typedef __attribute__((ext_vector_type(8))) float v8f;

// ---- parameter pointer table (setup_inputs dict insertion order, pytree
// flattened: lists in order, tuples in order) ----
// 0: x (B,N,3)   1: A (B,N,N)
// 2..5   input : W(3,8) b g bt
// 6..17  vp    : 3 layers x (W(8,8) b g bt)
// 18..33 node  : layer0 W(32,8) b g bt ; layers1-3 W(8,8) b g bt
// 34..43 edge  : W0(25,8) b0 W1(8,8) b1 W2 b2 W3 b3 Wl(8,1) bl(1)
struct P44 { const float* p[44]; };

enum { P_X = 0, P_A = 1, P_IN = 2, P_VP = 6, P_NODE = 18, P_EDGE = 34 };

// ---------------- helpers ----------------
__device__ __forceinline__ void mlp_layer(const float* hin, int K, const float* W,
                                          const float* bb, const float* g,
                                          const float* bt, float* hout) {
  float t[HD];
#pragma unroll
  for (int m = 0; m < HD; ++m) {
    float s = bb[m];
    for (int k = 0; k < K; ++k) s += hin[k] * W[k * HD + m];
    t[m] = s;
  }
  float mu = 0.f;
#pragma unroll
  for (int m = 0; m < HD; ++m) mu += t[m];
  mu *= 0.125f;
  float var = 0.f;
#pragma unroll
  for (int m = 0; m < HD; ++m) { float d = t[m] - mu; var += d * d; }
  var *= 0.125f;
  float inv = rsqrtf(var + 1e-5f);
#pragma unroll
  for (int m = 0; m < HD; ++m) hout[m] = tanhf((t[m] - mu) * inv * g[m] + bt[m]);
}

// ---------------- stage 1: input MLP (3 -> 8, LN + tanh) ----------------
__global__ void k_input_mlp(const float* __restrict__ x, P44 pp,
                            float* __restrict__ out) {
  int n = blockIdx.x * blockDim.x + threadIdx.x;
  if (n >= NNODES) return;
  const float* W = pp.p[P_IN + 0];
  const float* b = pp.p[P_IN + 1];
  const float* g = pp.p[P_IN + 2];
  const float* bt = pp.p[P_IN + 3];
  float xi[3];
#pragma unroll
  for (int d = 0; d < 3; ++d) xi[d] = x[n * 3 + d];
  float t[HD];
#pragma unroll
  for (int m = 0; m < HD; ++m) {
    float s = b[m];
#pragma unroll
    for (int d = 0; d < 3; ++d) s += xi[d] * W[d * HD + m];
    t[m] = s;
  }
  float mu = 0.f;
#pragma unroll
  for (int m = 0; m < HD; ++m) mu += t[m];
  mu *= 0.125f;
  float var = 0.f;
#pragma unroll
  for (int m = 0; m < HD; ++m) { float d = t[m] - mu; var += d * d; }
  var *= 0.125f;
  float inv = rsqrtf(var + 1e-5f);
#pragma unroll
  for (int m = 0; m < HD; ++m)
    out[n * HD + m] = tanhf((t[m] - mu) * inv * g[m] + bt[m]);
}

// ---------------- stage 2: mean over nodes + vp MLP (3 layers 8->8) ----------
__global__ void k_mean_vp(const float* __restrict__ xin, P44 pp,
                          float* __restrict__ vp) {
  __shared__ float acc[HD];
  int b = blockIdx.x;
  int t = threadIdx.x;
  if (t < HD) acc[t] = 0.f;
  __syncthreads();
  float loc[HD] = {0.f, 0.f, 0.f, 0.f, 0.f, 0.f, 0.f, 0.f};
  for (int nn = t; nn < NN; nn += blockDim.x) {
    const float* r = xin + (size_t)(b * NN + nn) * HD;
#pragma unroll
    for (int m = 0; m < HD; ++m) loc[m] += r[m];
  }
#pragma unroll
  for (int m = 0; m < HD; ++m) atomicAdd(&acc[m], loc[m]);
  __syncthreads();
  if (t == 0) {
    float h[HD], h2[HD];
#pragma unroll
    for (int m = 0; m < HD; ++m) h[m] = acc[m] * (1.f / (float)NN);
#pragma unroll
    for (int L = 0; L < 3; ++L) {
      mlp_layer(h, HD, pp.p[P_VP + L * 4 + 0], pp.p[P_VP + L * 4 + 1],
                pp.p[P_VP + L * 4 + 2], pp.p[P_VP + L * 4 + 3], h2);
#pragma unroll
      for (int m = 0; m < HD; ++m) h[m] = h2[m];
    }
#pragma unroll
    for (int m = 0; m < HD; ++m) vp[b * HD + m] = h[m];
  }
}

// ---------------- stage 3a: mi = A^T x, mo = A x  via WMMA f32 16x16x4 -------
// 16 blocks per batch; each block stages x8[b] (32 KB) in LDS, then its 8
// waves each run one (mat, 16-row-tile) GEMM task with a K=1024 loop.
__global__ void k_node_agg(const float* __restrict__ A,
                           const float* __restrict__ x8,
                           float* __restrict__ mi, float* __restrict__ mo) {
  __shared__ float xs[NN * HD];  // 32 KB of the 320 KB WGP LDS
  int b = blockIdx.x >> 4;
  int bb = blockIdx.x & 15;
  const float* xb = x8 + (size_t)b * NN * HD;
  for (int idx = threadIdx.x; idx < NN * HD; idx += blockDim.x)
    xs[idx] = xb[idx];
  __syncthreads();

  int wave = threadIdx.x >> 5;
  int lane = threadIdx.x & 31;
  int half = lane >> 4;
  int l = lane & 15;
  int task = bb * 8 + wave;  // 0..127
  int mat = task >> 6;       // 0 = mo (A[i][j]), 1 = mi (A^T)
  int tile = task & 63;
  int i0 = tile * 16;
  const float* Ab = A + (size_t)b * NN * NN;
  int n = l;
  bool nvalid = n < HD;
  int nc = n & 7;  // clamped column so loads are unconditional
  size_t abase = mat ? (size_t)(i0 + l) : (size_t)(i0 + l) * NN;
  size_t sk = mat ? (size_t)NN : (size_t)1;
  v8f acc = {};
#pragma unroll 4
  for (int k0 = 0; k0 < NN; k0 += 4) {
    int ka = k0 + half * 2;
    v2f av, bv;
    av.x = Ab[abase + (size_t)ka * sk];
    av.y = Ab[abase + (size_t)(ka + 1) * sk];
    float vx0 = xs[ka * HD + nc];
    float vx1 = xs[(ka + 1) * HD + nc];
    bv.x = nvalid ? vx0 : 0.f;
    bv.y = nvalid ? vx1 : 0.f;
    acc = __builtin_amdgcn_wmma_f32_16x16x4_f32(false, av, false, bv, (short)0,
                                                acc, false, false);
  }
  float* dst = (mat ? mi : mo) + (size_t)b * NN * HD;
  if (nvalid) {
#pragma unroll
    for (int r = 0; r < 8; ++r)
      dst[(size_t)(i0 + r + half * 8) * HD + n] = acc[r];
  }
}

// ---------------- stage 3b: node MLP (concat 32 -> 8 -> 8 -> 8 -> 8) --------
__global__ void k_node_mlp(const float* __restrict__ mi,
                           const float* __restrict__ mo,
                           const float* __restrict__ xc,
                           const float* __restrict__ vp, P44 pp,
                           float* __restrict__ xout) {
  int n = blockIdx.x * blockDim.x + threadIdx.x;
  if (n >= NNODES) return;
  int b = n >> 10;
  float feat[32];
#pragma unroll
  for (int m = 0; m < HD; ++m) {
    feat[m] = mi[n * HD + m];
    feat[8 + m] = mo[n * HD + m];
    feat[16 + m] = xc[n * HD + m];
    feat[24 + m] = vp[b * HD + m];
  }
  float h[HD], h2[HD];
  mlp_layer(feat, 32, pp.p[P_NODE + 0], pp.p[P_NODE + 1], pp.p[P_NODE + 2],
            pp.p[P_NODE + 3], h);
#pragma unroll
  for (int L = 1; L < 4; ++L) {
    mlp_layer(h, HD, pp.p[P_NODE + L * 4 + 0], pp.p[P_NODE + L * 4 + 1],
              pp.p[P_NODE + L * 4 + 2], pp.p[P_NODE + L * 4 + 3], h2);
#pragma unroll
    for (int m = 0; m < HD; ++m) h[m] = h2[m];
  }
#pragma unroll
  for (int m = 0; m < HD; ++m) xout[n * HD + m] = h[m];
}

// ---------------- stage 4a: per-node edge precompute ------------------------
// cj = x@Wj (W0 rows 0..7), di = x@Wi (rows 8..15), base = vp@Wv + b0 (rows 16..23)
__global__ void k_edge_pre(const float* __restrict__ xt,
                           const float* __restrict__ vp, P44 pp,
                           float* __restrict__ cj, float* __restrict__ di,
                           float* __restrict__ base) {
  int n = blockIdx.x * blockDim.x + threadIdx.x;
  if (n >= NNODES) return;
  const float* W0 = pp.p[P_EDGE + 0];
  const float* b0 = pp.p[P_EDGE + 1];
  const float* xv = xt + (size_t)n * HD;
#pragma unroll
  for (int m = 0; m < HD; ++m) {
    float sj = 0.f, si = 0.f;
#pragma unroll
    for (int k = 0; k < HD; ++k) {
      sj += xv[k] * W0[k * HD + m];
      si += xv[k] * W0[(8 + k) * HD + m];
    }
    cj[n * HD + m] = sj;
    di[n * HD + m] = si;
  }
  if ((n & (NN - 1)) == 0) {
    int b = n >> 10;
#pragma unroll
    for (int m = 0; m < HD; ++m) {
      float s = b0[m];
#pragma unroll
      for (int k = 0; k < HD; ++k) s += vp[b * HD + k] * W0[(16 + k) * HD + m];
      base[b * HD + m] = s;
    }
  }
}

// ---------------- stage 4b: edge net, WMMA-batched over 16 edges ------------
// one block per (b,i) row; each wave handles 16-j chunks. Hidden vectors are
// kept in native B-fragment layout: lane<16 owns h[0,1,4,5] of edge (lane&15),
// lane>=16 owns h[2,3,6,7]. A-fragments are W^T padded to 16 rows.
__global__ void k_edge(const float* __restrict__ A_in, float* __restrict__ A_out,
                       const float* __restrict__ cj, const float* __restrict__ di,
                       const float* __restrict__ base, P44 pp) {
  int row = blockIdx.x;  // 0 .. B*N-1
  int b = row >> 10;
  int i = row & (NN - 1);
  int wave = threadIdx.x >> 5;
  int lane = threadIdx.x & 31;
  int half = lane >> 4;
  int l = lane & 15;
  int c0 = half * 2;  // this lane's k base: {0,1,4,5} or {2,3,6,7}

  const float* W0 = pp.p[P_EDGE + 0];
  const float* wA = W0 + 24 * HD;

  const float* dip = di + (size_t)(b * NN + i) * HD;
  float diA = dip[c0], diB = dip[c0 + 1], diC = dip[c0 + 4], diD = dip[c0 + 5];
  float baA = base[b * HD + c0], baB = base[b * HD + c0 + 1];
  float baC = base[b * HD + c0 + 4], baD = base[b * HD + c0 + 5];
  float wAA = wA[c0], wAB = wA[c0 + 1], wAC = wA[c0 + 4], wAD = wA[c0 + 5];

  // loop-invariant A-fragments for the three 8x8 layers (Amat[m][k] = W[k][m])
  v2f af[3][2];
  float bias[3][HD];
#pragma unroll
  for (int L = 0; L < 3; ++L) {
    const float* W = pp.p[P_EDGE + 2 + L * 2];
    const float* bv = pp.p[P_EDGE + 3 + L * 2];
#pragma unroll
    for (int cc = 0; cc < 2; ++cc) {
      int kk = cc * 4 + c0;
      af[L][cc].x = (l < 8) ? W[kk * HD + l] : 0.f;
      af[L][cc].y = (l < 8) ? W[(kk + 1) * HD + l] : 0.f;
    }
#pragma unroll
    for (int m = 0; m < HD; ++m) bias[L][m] = bv[m];
  }
  const float* Wl = pp.p[P_EDGE + 8];
  float wlA = Wl[c0], wlB = Wl[c0 + 1], wlC = Wl[c0 + 4], wlD = Wl[c0 + 5];
  float blv = pp.p[P_EDGE + 9][0];

  const float* Arow = A_in + (size_t)b * NN * NN + (size_t)i * NN;
  float* Orow = A_out + (size_t)b * NN * NN + (size_t)i * NN;

  for (int cg = wave; cg < NN / 16; cg += 8) {
    int j0 = cg * 16;
    int j = j0 + l;
    // prefetch next chunk of this row into cache (speculative)
    __builtin_prefetch(Arow + ((cg + 8) & (NN / 16 - 1)) * 16, 0, 1);

    const float* cjp = cj + (size_t)(b * NN + j) * HD;
    float2 c01 = *(const float2*)(cjp + c0);
    float2 c45 = *(const float2*)(cjp + c0 + 4);
    float a = Arow[j];
    float hA = tanhf(c01.x + diA + baA + a * wAA);
    float hB = tanhf(c01.y + diB + baB + a * wAB);
    float hC = tanhf(c45.x + diC + baC + a * wAC);
    float hD = tanhf(c45.y + diD + baD + a * wAD);

#pragma unroll
    for (int L = 0; L < 3; ++L) {
      v8f acc = {};
      v2f b0v; b0v.x = hA; b0v.y = hB;
      v2f b1v; b1v.x = hC; b1v.y = hD;
      acc = __builtin_amdgcn_wmma_f32_16x16x4_f32(false, af[L][0], false, b0v,
                                                  (short)0, acc, false, false);
      acc = __builtin_amdgcn_wmma_f32_16x16x4_f32(false, af[L][1], false, b1v,
                                                  (short)0, acc, false, false);
      float p0 = acc[0] + bias[L][0], p1 = acc[1] + bias[L][1];
      float p2 = acc[2] + bias[L][2], p3 = acc[3] + bias[L][3];
      float p4 = acc[4] + bias[L][4], p5 = acc[5] + bias[L][5];
      float p6 = acc[6] + bias[L][6], p7 = acc[7] + bias[L][7];
      // D -> B relayout: upper lane-half pulls rows 2,3,6,7 from lanes 0..15
      float s2 = __shfl(p2, l), s3 = __shfl(p3, l);
      float s6 = __shfl(p6, l), s7 = __shfl(p7, l);
      hA = tanhf(half ? s2 : p0);
      hB = tanhf(half ? s3 : p1);
      hC = tanhf(half ? s6 : p4);
      hD = tanhf(half ? s7 : p5);
    }
    // final 8 -> 1 + sigmoid
    float part = hA * wlA + hB * wlB + hC * wlC + hD * wlD;
    part += __shfl_xor(part, 16);
    float val = part + blv;
    float sg = 1.0f / (1.0f + __expf(-val));
    if (!half) Orow[j] = sg;
  }
}

// ---------------- residual / copy ----------------
__global__ void k_residual(float* __restrict__ xcur,
                           const float* __restrict__ xtmp, int n) {
  int i = blockIdx.x * blockDim.x + threadIdx.x;
  if (i < n) xcur[i] = xcur[i] + xtmp[i];
}

__global__ void k_copy(const float* __restrict__ src, float* __restrict__ dst,
                       int n) {
  int i = blockIdx.x * blockDim.x + threadIdx.x;
  if (i < n) dst[i] = src[i];
}

// ---------------- host launch ----------------
extern "C" void kernel_launch(void* const* d_in, const int* in_sizes, int n_in,
                              void* d_out, int out_size, void* d_ws,
                              size_t ws_size, hipStream_t stream) {
  (void)in_sizes; (void)out_size; (void)ws_size;
  P44 pp;
  for (int i = 0; i < 44; ++i)
    pp.p[i] = (const float*)d_in[(i < n_in) ? i : 0];

  const float* x_in = pp.p[P_X];
  const float* A_in0 = pp.p[P_A];
  float* out = (float*)d_out;

  float* ws = (float*)d_ws;
  float* x_cur = ws;                 // 16384
  float* x_tmp = ws + 16384;         // 16384
  float* mi = ws + 32768;            // 16384
  float* mo = ws + 49152;            // 16384
  float* cj = ws + 65536;            // 16384
  float* di = ws + 81920;            // 16384
  float* vp = ws + 98304;            // 16 (padded)
  float* base = ws + 98368;          // 16 (padded)
  float* A_ws = ws + 98432;          // 2M
  float* A_dout = out + NNODES * HD; // final A region in d_out

  k_input_mlp<<<NNODES / 256, 256, 0, stream>>>(x_in, pp, x_cur);
  k_mean_vp<<<BATCH, 256, 0, stream>>>(x_cur, pp, vp);

  const float* Acur = A_in0;
  float* Anext = A_dout;
  for (int it = 0; it < 3; ++it) {
    k_node_agg<<<32, 256, 0, stream>>>(Acur, x_cur, mi, mo);
    k_node_mlp<<<NNODES / 256, 256, 0, stream>>>(mi, mo, x_cur, vp, pp, x_tmp);
    k_edge_pre<<<NNODES / 256, 256, 0, stream>>>(x_tmp, vp, pp, cj, di, base);
    k_edge<<<NNODES, 256, 0, stream>>>(Acur, Anext, cj, di, base, pp);
    k_mean_vp<<<BATCH, 256, 0, stream>>>(x_tmp, pp, vp);
    k_residual<<<NNODES * HD / 256, 256, 0, stream>>>(x_cur, x_tmp, NNODES * HD);
    if (it == 0) { Acur = A_dout; Anext = A_ws; }
    else if (it == 1) { Acur = A_ws; Anext = A_dout; }
  }
  k_copy<<<NNODES * HD / 256, 256, 0, stream>>>(x_cur, out, NNODES * HD);
}